// RNN_9259949490669
// MI455X (gfx1250) — compile-verified
//
#include <hip/hip_runtime.h>

typedef __attribute__((ext_vector_type(2))) float v2f;
typedef __attribute__((ext_vector_type(8))) float v8f;

#define B_   64
#define T_   1024
#define D_   512
#define U_   1024
#define G_   4096          // 4*U
#define K_   1536          // D + U
#define RS   1540          // padded LDS row stride (dwords): 16B-aligned rows, bank-conflict free
#define BT   16            // b rows per workgroup
#define NWGU 16            // U / 64
#define NWGB 4             // B / 16
#define NWG  (NWGB * NWGU) // 64 persistent workgroups
#define LDS_BYTES (BT * RS * 4)

__device__ __forceinline__ float rcpf(float x) { return __builtin_amdgcn_rcpf(x); }
__device__ __forceinline__ float sigm(float x) {
    return rcpf(1.0f + __expf(-x));            // v_exp_f32 + v_rcp_f32, no div-fixup
}
__device__ __forceinline__ float tanh_f(float x) {
    float e = __expf(2.0f * x);
    return 1.0f - 2.0f * rcpf(e + 1.0f);       // -> -1 / +1 at extremes, no NaN
}

// async global->LDS copy of 16 bytes, per-lane addresses (CDNA5 ASYNCcnt path)
__device__ __forceinline__ void async_b128(const void* gaddr, void* lds_ptr) {
    unsigned ldsa = (unsigned)(uintptr_t)lds_ptr;   // low 32 bits = LDS byte address
    asm volatile("global_load_async_to_lds_b128 %0, %1, off"
                 :: "v"(ldsa), "v"(gaddr) : "memory");
}
__device__ __forceinline__ void wait_async0() {
    asm volatile("s_wait_asynccnt 0x0" ::: "memory");
}

extern "C" __global__ void lstm_init(unsigned* cnt, float* h0) {
    int i = blockIdx.x * blockDim.x + threadIdx.x;
    if (i == 0) *cnt = 0u;
    if (i < B_ * U_) h0[i] = 0.0f;
}

extern "C" __global__ void __launch_bounds__(128, 1)
lstm_persist(const float* __restrict__ x, const float* __restrict__ W,
             const float* __restrict__ bias, float* __restrict__ out,
             float* __restrict__ hbuf0, float* __restrict__ hbuf1,
             unsigned* __restrict__ cnt)
{
    extern __shared__ float lds[];
    const int tid  = threadIdx.x;
    const int wave = tid >> 5;
    const int lane = tid & 31;
    const int m16  = lane & 15;           // A row / C col within tile
    const int kk   = (lane >> 4) << 1;    // K sub-offset: 0 or 2
    const int wg   = blockIdx.x;
    const int b0   = (wg / NWGU) * BT;    // b-row base
    const int u0   = (wg % NWGU) * 64 + wave * 16;  // this wave's u-col base
    const int n    = m16;

    // Per-gate weight column pointers (L2-resident) and bias values
    const float* wp[4];
    float bb[4];
#pragma unroll
    for (int g = 0; g < 4; ++g) {
        wp[g] = W + (size_t)(g * U_ + u0 + n);
        bb[g] = bias[g * U_ + u0 + n];
    }

    v8f cstate = {};                       // cell-state tile lives in VGPRs
    float* hb[2] = { hbuf0, hbuf1 };

    for (int t = 0; t < T_; ++t) {
        const float* hread  = hb[t & 1];
        float*       hwrite = hb[(t + 1) & 1];

        // ---- stage A = [x_t | h_{t-1}] (16 x 1536) into LDS, async direct-to-LDS ----
        for (int i = tid; i < BT * (D_ / 4); i += 128) {
            int m = i / (D_ / 4), c = i % (D_ / 4);
            async_b128(x + ((size_t)(b0 + m) * T_ + t) * D_ + 4 * c,
                       &lds[m * RS + 4 * c]);
        }
        for (int i = tid; i < BT * (U_ / 4); i += 128) {
            int m = i / (U_ / 4), c = i % (U_ / 4);
            async_b128(hread + (size_t)(b0 + m) * U_ + 4 * c,
                       &lds[m * RS + D_ + 4 * c]);
        }
        wait_async0();
        __syncthreads();

        // ---- gates(16x64) = A(16x1536) @ W(1536 x 4 gate tiles), fp32 WMMA ----
        v8f acc[4] = {};
#pragma unroll 4
        for (int k = 0; k < K_; k += 4) {
            v2f a = *(const v2f*)&lds[m16 * RS + k + kk];   // conflict-free LDS read
#pragma unroll
            for (int g = 0; g < 4; ++g) {
                v2f bf;
                bf.x = wp[g][(size_t)(k + kk)     * G_];
                bf.y = wp[g][(size_t)(k + kk + 1) * G_];
                acc[g] = __builtin_amdgcn_wmma_f32_16x16x4_f32(
                    false, a, false, bf, (short)0, acc[g], false, false);
            }
        }
        __syncthreads();   // all waves done reading LDS before next-step staging

        // ---- LSTM cell elementwise (acc row m = r + 8*(lane>>4), col n) ----
        v8f hnew;
#pragma unroll
        for (int r = 0; r < 8; ++r) {
            float gi = acc[0][r] + bb[0];
            float gj = acc[1][r] + bb[1];
            float gf = acc[2][r] + bb[2];
            float go = acc[3][r] + bb[3];
            float nc = cstate[r] * sigm(gf + 1.0f) + sigm(gi) * tanh_f(gj);
            cstate[r] = nc;
            hnew[r]   = tanh_f(nc) * sigm(go);
        }

        const int mbase = (lane >> 4) * 8;
#pragma unroll
        for (int r = 0; r < 8; ++r) {
            int m = mbase + r;
            float v = hnew[r];
            hwrite[(size_t)(b0 + m) * U_ + u0 + n]          = v;
            out[((size_t)(b0 + m) * T_ + t) * U_ + u0 + n]  = v;
        }

        // ---- device-wide barrier: all h writes visible before next step ----
        __threadfence();
        __syncthreads();
        if (tid == 0) {
            atomicAdd(cnt, 1u);
            unsigned target = (unsigned)(t + 1) * NWG;
            while (__hip_atomic_load(cnt, __ATOMIC_RELAXED,
                                     __HIP_MEMORY_SCOPE_AGENT) < target) {
                __builtin_amdgcn_s_sleep(1);
            }
        }
        __syncthreads();
        __threadfence();
    }
}

extern "C" void kernel_launch(void* const* d_in, const int* in_sizes, int n_in,
                              void* d_out, int out_size, void* d_ws, size_t ws_size,
                              hipStream_t stream) {
    const float* xin  = (const float*)d_in[0];   // (B,T,D) f32
    const float* Wk   = (const float*)d_in[1];   // (D+U, 4U) f32
    const float* bias = (const float*)d_in[2];   // (4U,) f32
    float* out = (float*)d_out;                  // (B,T,U) f32

    unsigned char* ws = (unsigned char*)d_ws;
    unsigned* cnt = (unsigned*)ws;               // barrier counter
    float* h0 = (float*)(ws + 1024);             // h ping buffer (256 KB)
    float* h1 = h0 + B_ * U_;                    // h pong buffer (256 KB)

    lstm_init<<<(B_ * U_ + 255) / 256, 256, 0, stream>>>(cnt, h0);
    lstm_persist<<<NWG, 128, LDS_BYTES, stream>>>(xin, Wk, bias, out, h0, h1, cnt);
}